// CRF_5196910428677
// MI455X (gfx1250) — compile-verified
//
#include <hip/hip_runtime.h>

// CRF negative log-likelihood, fused single-pass kernel for gfx1250 (MI455X).
// - Emission GEMM (X . W^T) and the exp-space forward recursion (U . exp(T))
//   both run on v_wmma_f32_16x16x4_f32 (memory-bound problem -> f32 WMMA,
//   full precision, compute nowhere near the ceiling).
// - X tiles are streamed with double-buffered GLOBAL_LOAD_ASYNC_TO_LDS_B128
//   (ASYNCcnt), overlapping the per-timestep 2KB-per-wave fetch with the WMMA phase.

typedef float v2f __attribute__((ext_vector_type(2)));
typedef float v8f __attribute__((ext_vector_type(8)));

#define BATCH    8192
#define MLEN     20
#define KLAB     26     // labels
#define DIM      128    // feature dim
#define XPAD     132    // LDS row stride for 128-wide rows (bank-conflict free)
#define APAD     34     // LDS row stride for 32-wide rows (bank-conflict free)
#define WPB      2      // waves per block
#define NB       16     // batches per wave (== WMMA M)

__device__ __forceinline__ void async_load_b128(uint32_t lds_off, const float* gp) {
  asm volatile("global_load_async_to_lds_b128 %0, %1, off"
               :: "v"(lds_off), "v"((unsigned long long)(uintptr_t)gp)
               : "memory");
}
__device__ __forceinline__ void wait_async_le4() {
  asm volatile("s_wait_asynccnt 0x4" ::: "memory");
}
__device__ __forceinline__ void wait_async_0() {
  asm volatile("s_wait_asynccnt 0x0" ::: "memory");
}

__global__ __launch_bounds__(WPB * 32)
void crf_fused_kernel(const float* __restrict__ Xg,
                      const int*   __restrict__ Yg,
                      const float* __restrict__ Wg,
                      const float* __restrict__ Tg,
                      float* __restrict__ partial) {
  __shared__ __align__(16) float Wl[KLAB][XPAD];          // W rows 0..25
  __shared__ __align__(16) float Tl[KLAB][KLAB];          // raw T (edge potentials)
  __shared__ __align__(16) float Xt[WPB][2][NB][XPAD];    // double-buffered X tile
  __shared__ __align__(16) float El[WPB][NB][APAD];       // emission tile per wave
  __shared__ __align__(16) float Ul[WPB][NB][APAD];       // U matrix (A of recursion)
  __shared__ __align__(16) float Au[WPB][NB][APAD];       // araw (exp-space alpha)

  const int tid    = threadIdx.x;
  const int wave   = tid >> 5;
  const int lane   = tid & 31;
  const int laneN  = lane & 15;
  const int laneHi = lane >> 4;
  const int ak     = laneHi * 2;      // K sub-offset inside a 16x16x4 fragment

  const int b0 = blockIdx.x * (WPB * NB) + wave * NB;   // first global batch of wave

  // LDS byte offsets of the two X buffers (generic LDS addr low 32 bits == offset)
  const uint32_t xbase0 = (uint32_t)(uintptr_t)&Xt[wave][0][0][0];
  const uint32_t xbase1 = (uint32_t)(uintptr_t)&Xt[wave][1][0][0];

  // ---- kick off async staging of timestep 0 (4 x b128, 4 rows per instr) ----
  const int rr = lane >> 3;           // row-in-group
  const int cc = lane & 7;            // 16B chunk within a 512B row
#pragma unroll
  for (int g = 0; g < 4; ++g) {
    int r = g * 4 + rr;
    async_load_b128(xbase0 + (uint32_t)(r * XPAD * 4 + cc * 16),
                    Xg + ((size_t)(b0 + r) * MLEN + 0) * DIM + cc * 4);
  }

  // ---- stage W (rows 0..25) and raw T ----
  for (int i = tid; i < KLAB * DIM; i += WPB * 32) {
    int n = i >> 7, c = i & 127;
    Wl[n][c] = Wg[n * DIM + c];
  }
  for (int i = tid; i < KLAB * KLAB; i += WPB * 32)
    Tl[i / KLAB][i % KLAB] = Tg[i];

  // ---- init recursion state: alpha0 = 0  =>  araw = 1 (k<26), 0 pad ----
  for (int i = lane; i < NB * APAD; i += 32) {
    int c = i % APAD;
    Ul[wave][i / APAD][c] = 0.0f;
    Au[wave][i / APAD][c] = (c < KLAB) ? 1.0f : 0.0f;
  }
  __syncthreads();

  // ---- preload exp(T) B-fragments for the recursion matmul (held in VGPRs) ----
  v2f bt[8][2];
#pragma unroll
  for (int ks = 0; ks < 8; ++ks) {
#pragma unroll
    for (int nt = 0; nt < 2; ++nt) {
      int kk = ks * 4 + ak;
      int n  = laneN + nt * 16;
      v2f bv;
      bv.x = (kk     < KLAB && n < KLAB) ? __expf(Tl[kk][n])     : 0.0f;
      bv.y = (kk + 1 < KLAB && n < KLAB) ? __expf(Tl[kk + 1][n]) : 0.0f;
      bt[ks][nt] = bv;
    }
  }

  // emission B-fragment rows (clamp dead padded labels 26..31 into range; the
  // resulting D columns 26..31 are never read)
  const int wr0 = laneN;
  const int wr1 = (laneN + 16 < KLAB) ? (laneN + 16) : laneN;

  const int bl  = lane >> 1;          // batch for elementwise phases
  const int sub = lane & 1;           // half of label range (13 each)

  float c_acc = 0.0f;   // running log-scale of the exp-space alpha (per pair)
  float node  = 0.0f;   // accumulated on sub==0 lanes
  float edge  = 0.0f;   // accumulated on sub==1 lanes
  float res   = 0.0f;

  for (int s = 0; s < MLEN; ++s) {
    const int buf = s & 1;

    // -------- issue async staging for s+1 into the other buffer, then wait
    // -------- for this step's 4 loads (async loads complete in order) --------
    if (s + 1 < MLEN) {
      uint32_t nb = (buf ? xbase0 : xbase1);
#pragma unroll
      for (int g = 0; g < 4; ++g) {
        int r = g * 4 + rr;
        async_load_b128(nb + (uint32_t)(r * XPAD * 4 + cc * 16),
                        Xg + ((size_t)(b0 + r) * MLEN + (s + 1)) * DIM + cc * 4);
      }
      wait_async_le4();
    } else {
      wait_async_0();
    }
    // Xt is wave-private: no barrier needed, the asynccnt wait is sufficient.

    // -------- emission GEMM: D(16x32) = Xt(16x128) * W^T, f32 WMMA --------
    v8f acc0 = {};
    v8f acc1 = {};
#pragma unroll
    for (int k = 0; k < DIM; k += 4) {
      v2f a   = *(const v2f*)&Xt[wave][buf][laneN][k + ak];
      v2f bb0 = *(const v2f*)&Wl[wr0][k + ak];        // B[k][n] = W[n][k]
      v2f bb1 = *(const v2f*)&Wl[wr1][k + ak];
      acc0 = __builtin_amdgcn_wmma_f32_16x16x4_f32(false, a, false, bb0,
                                                   (short)0, acc0, false, false);
      acc1 = __builtin_amdgcn_wmma_f32_16x16x4_f32(false, a, false, bb1,
                                                   (short)0, acc1, false, false);
    }
    // spill emission tile to LDS (D layout: VGPR r -> row, lane half -> +8)
#pragma unroll
    for (int r = 0; r < 8; ++r) {
      El[wave][laneHi * 8 + r][laneN]      = acc0[r];
      El[wave][laneHi * 8 + r][laneN + 16] = acc1[r];
    }
    __syncthreads();

    // -------- node / edge potential gathers --------
    int y_s = Yg[(size_t)(b0 + bl) * MLEN + s];
    if (sub == 0) {
      node += El[wave][bl][y_s];
    } else if (s + 1 < MLEN) {
      int y_n = Yg[(size_t)(b0 + bl) * MLEN + s + 1];
      edge += Tl[y_s][y_n];
    }

    // -------- exp-space U-build: u = (araw/mA) * exp(e - me) --------
    float ar[13], ev[13];
    float mA = 1e-30f, me = -3.0e38f;
    const int kb = sub * 13;
#pragma unroll
    for (int t = 0; t < 13; ++t) {
      ar[t] = Au[wave][bl][kb + t];
      ev[t] = El[wave][bl][kb + t];
      mA = fmaxf(mA, ar[t]);
      me = fmaxf(me, ev[t]);
    }
    mA = fmaxf(mA, __shfl_xor(mA, 1, 32));
    me = fmaxf(me, __shfl_xor(me, 1, 32));
    const float invA = 1.0f / mA;
    float usum = 0.0f;
#pragma unroll
    for (int t = 0; t < 13; ++t) {
      float u = ar[t] * invA * __expf(ev[t] - me);
      usum += u;
      if (s + 1 < MLEN) Ul[wave][bl][kb + t] = u;
    }
    c_acc += me + __logf(mA);

    if (s + 1 < MLEN) {
      __syncthreads();
      // -------- recursion matmul: araw' = U(16x32) * expT(32x32), f32 WMMA ----
      v8f r0 = {};
      v8f r1 = {};
#pragma unroll
      for (int ks = 0; ks < 8; ++ks) {
        v2f a = *(const v2f*)&Ul[wave][laneN][ks * 4 + ak];
        r0 = __builtin_amdgcn_wmma_f32_16x16x4_f32(false, a, false, bt[ks][0],
                                                   (short)0, r0, false, false);
        r1 = __builtin_amdgcn_wmma_f32_16x16x4_f32(false, a, false, bt[ks][1],
                                                   (short)0, r1, false, false);
      }
#pragma unroll
      for (int r = 0; r < 8; ++r) {
        Au[wave][laneHi * 8 + r][laneN]      = r0[r];
        Au[wave][laneHi * 8 + r][laneN + 16] = r1[r];
      }
      __syncthreads();
    } else {
      // -------- finalize: log Z = c + log(sum_k u_k), NLL contribution --------
      usum += __shfl_xor(usum, 1, 32);
      float logz = c_acc + __logf(usum);
      float mine  = (sub == 0) ? node : edge;
      float other = __shfl_xor(mine, 1, 32);
      res = (sub == 0) ? -(mine + other - logz) : 0.0f;
    }
  }

  // deterministic wave reduction of the 16 per-batch NLL values
#pragma unroll
  for (int off = 16; off > 0; off >>= 1)
    res += __shfl_xor(res, off, 32);
  if (lane == 0)
    partial[blockIdx.x * WPB + wave] = res;
}

__global__ __launch_bounds__(512)
void crf_reduce_kernel(const float* __restrict__ partial, float* __restrict__ out,
                       int n) {
  __shared__ float sm[512];
  int t = threadIdx.x;
  sm[t] = (t < n) ? partial[t] : 0.0f;
  __syncthreads();
  for (int off = 256; off > 0; off >>= 1) {
    if (t < off) sm[t] += sm[t + off];
    __syncthreads();
  }
  if (t == 0) out[0] = sm[0];
}

extern "C" void kernel_launch(void* const* d_in, const int* in_sizes, int n_in,
                              void* d_out, int out_size, void* d_ws, size_t ws_size,
                              hipStream_t stream) {
  (void)in_sizes; (void)n_in; (void)out_size; (void)ws_size;
  const float* X = (const float*)d_in[0];
  const int*   Y = (const int*)d_in[1];
  const float* W = (const float*)d_in[2];
  const float* T = (const float*)d_in[3];
  float* out     = (float*)d_out;
  float* partial = (float*)d_ws;   // 512 floats of scratch

  const int nblocks = BATCH / (WPB * NB);   // 256
  crf_fused_kernel<<<nblocks, WPB * 32, 0, stream>>>(X, Y, W, T, partial);
  crf_reduce_kernel<<<1, 512, 0, stream>>>(partial, out, nblocks * WPB);
}